// ParallelMambaBlock_82016695485377
// MI455X (gfx1250) — compile-verified
//
#include <hip/hip_runtime.h>

// ---------------------------------------------------------------------------
// Mamba block forward for MI455X (gfx1250, wave32).
// GEMMs use V_WMMA_F32_16X16X4_F32 (fp32 matrix pipe, exact fp32 semantics).
// 128x128 block tile, 8 waves in a 4x2 grid, 32x64 per wave (8 accumulators):
// ds:wmma ratio 6:8 per K-step (vs 5:4 before). Global->LDS staging is
// software-pipelined through registers to overlap with WMMA work.
// Selective scan uses a 3-pass chunked parallel scan (16 chunks of 128).
// ---------------------------------------------------------------------------

#define L_SEQ   2048
#define N_BATCH 2
#define D_MODEL 1024
#define D_INNER 2048
#define DT_RANK 64
#define D_STATE 16
#define NTOK    (L_SEQ * N_BATCH)       // 4096 tokens, row t = l*B + b
#define XPROJ_N (DT_RANK + 2 * D_STATE) // 96

#define NCHUNK  16
#define CLEN    (L_SEQ / NCHUNK)        // 128

// GEMM tiling
#define GBM 128
#define GBN 128
#define GBK 32
#define LDW (GBK + 4)                   // 36-float rows: 144B (16B aligned, conflict-free)

typedef float v2f __attribute__((ext_vector_type(2)));
typedef float v8f __attribute__((ext_vector_type(8)));

// ---------------------------------------------------------------------------
// Generic WMMA GEMM:  C[M,N] = A[M,K] * Bw[N,K]^T   (all row-major, fp32)
// Frag layout per CDNA5 ISA 7.12.2:
//   A 16x4:  lane m<16 -> (A[m][k0],A[m][k0+1]), lane m+16 -> (k0+2,k0+3)
//   B 4x16:  lane n<16 -> (B[k0][n],B[k0+1][n]), lane n+16 -> (k0+2,k0+3)
//   C/D:     VGPR v -> row v (lanes 0-15) / row v+8 (lanes 16-31), col = lane&15
// ---------------------------------------------------------------------------
__global__ __launch_bounds__(256)
void wmma_gemm_nt(const float* __restrict__ A, const float* __restrict__ Bw,
                  float* __restrict__ C, int M, int N, int K,
                  int lda, int ldb, int ldc)
{
    __shared__ float Asm[GBM][LDW];
    __shared__ float Bsm[GBN][LDW];

    const int tid  = threadIdx.x;
    const int lane = tid & 31;
    const int wav  = tid >> 5;       // 0..7
    const int m0   = blockIdx.y * GBM;
    const int n0   = blockIdx.x * GBN;
    const int lr   = lane & 15;
    const int hi   = lane >> 4;      // half-wave -> K pair select
    const int wm   = (wav >> 1) * 32;   // wave row offset within tile (4 rows of waves)
    const int wn   = (wav & 1) * 64;    // wave col offset within tile (2 cols of waves)

    v8f acc[2][4] = {};
    float4 ra[4], rb[4];             // staging registers for the next K chunk

    auto load_regs = [&](int kc) {
        #pragma unroll
        for (int i = 0; i < 4; ++i) {
            int f  = tid + i * 256;          // 0..1023 slots (128 rows x 8 float4)
            int r  = f >> 3;
            int c4 = (f & 7) << 2;
            float4 va = {0.f, 0.f, 0.f, 0.f};
            float4 vb = {0.f, 0.f, 0.f, 0.f};
            if (m0 + r < M)
                va = *reinterpret_cast<const float4*>(A + (size_t)(m0 + r) * lda + kc + c4);
            if (n0 + r < N)
                vb = *reinterpret_cast<const float4*>(Bw + (size_t)(n0 + r) * ldb + kc + c4);
            ra[i] = va; rb[i] = vb;
        }
    };
    auto commit_lds = [&]() {
        #pragma unroll
        for (int i = 0; i < 4; ++i) {
            int f  = tid + i * 256;
            int r  = f >> 3;
            int c4 = (f & 7) << 2;
            *reinterpret_cast<float4*>(&Asm[r][c4]) = ra[i];
            *reinterpret_cast<float4*>(&Bsm[r][c4]) = rb[i];
        }
    };

    // prologue: stage chunk 0
    load_regs(0);
    commit_lds();

    for (int kc = 0; kc < K; kc += GBK) {
        __syncthreads();
        const bool more = (kc + GBK) < K;
        if (more) load_regs(kc + GBK);   // global loads overlap the WMMA loop below

        #pragma unroll
        for (int ks = 0; ks < GBK / 4; ++ks) {
            const int kk = ks * 4 + hi * 2;
            v2f a[2], b[4];
            #pragma unroll
            for (int mi = 0; mi < 2; ++mi) {
                a[mi][0] = Asm[wm + mi * 16 + lr][kk];
                a[mi][1] = Asm[wm + mi * 16 + lr][kk + 1];
            }
            #pragma unroll
            for (int ni = 0; ni < 4; ++ni) {
                b[ni][0] = Bsm[wn + ni * 16 + lr][kk];
                b[ni][1] = Bsm[wn + ni * 16 + lr][kk + 1];
            }
            #pragma unroll
            for (int mi = 0; mi < 2; ++mi)
                #pragma unroll
                for (int ni = 0; ni < 4; ++ni)
                    acc[mi][ni] = __builtin_amdgcn_wmma_f32_16x16x4_f32(
                        false, a[mi], false, b[ni], (short)0, acc[mi][ni], false, false);
        }
        __syncthreads();
        if (more) commit_lds();
    }

    #pragma unroll
    for (int mi = 0; mi < 2; ++mi) {
        const int rbase = m0 + wm + mi * 16 + hi * 8;
        #pragma unroll
        for (int ni = 0; ni < 4; ++ni) {
            int col = n0 + wn + ni * 16 + lr;
            if (col < N) {
                #pragma unroll
                for (int v = 0; v < 8; ++v)
                    C[(size_t)(rbase + v) * ldc + col] = acc[mi][ni][v];
            }
        }
    }
}

// ---------------------------------------------------------------------------
// Causal depthwise conv1d (width 4) + bias + SiLU.
// xz is token-major [NTOK][2*D_INNER]; x = cols [0,D_INNER).
// ---------------------------------------------------------------------------
__global__ __launch_bounds__(256)
void conv_silu_kernel(const float* __restrict__ xz, const float* __restrict__ cw,
                      const float* __restrict__ cb, float* __restrict__ x_act)
{
    int idx = blockIdx.x * 256 + threadIdx.x;      // over NTOK * D_INNER
    int c  = idx & (D_INNER - 1);
    int tb = idx >> 11;                            // token = l*B + b
    int b  = tb & (N_BATCH - 1);
    int l  = tb >> 1;
    float acc = cb[c];
    #pragma unroll
    for (int k = 0; k < 4; ++k) {
        int ls = l - 3 + k;
        if (ls >= 0)
            acc += cw[c * 4 + k] * xz[(size_t)(ls * N_BATCH + b) * (2 * D_INNER) + c];
    }
    x_act[(size_t)tb * D_INNER + c] = acc / (1.f + __expf(-acc));  // SiLU
}

__device__ __forceinline__ float softplusf(float x) {
    return (x > 20.f) ? x : log1pf(__expf(x));
}

// ---------------------------------------------------------------------------
// Scan pass 1: per (b, d, chunk) compute local end-state and cumulative dA
// product over the chunk (initial state 0).
// idx: d = low 11 bits, chunk = next 4, b = bit 15. Summary layout:
//   off(b,chunk,d) = ((b*NCHUNK + chunk)*D_INNER + d)*D_STATE
// ---------------------------------------------------------------------------
__global__ __launch_bounds__(256)
void scan_pass1(const float* __restrict__ dtm, const float* __restrict__ u_act,
                const float* __restrict__ x_dbl, const float* __restrict__ dt_bias,
                const float* __restrict__ A_log,
                float* __restrict__ stEnd, float* __restrict__ Ptot)
{
    int idx   = blockIdx.x * 256 + threadIdx.x;
    int d     = idx & (D_INNER - 1);
    int chunk = (idx >> 11) & (NCHUNK - 1);
    int b     = idx >> 15;

    float Arow[D_STATE];
    #pragma unroll
    for (int n = 0; n < D_STATE; ++n) Arow[n] = -__expf(A_log[d * D_STATE + n]);
    const float dtb = dt_bias[d];

    float st[D_STATE], P[D_STATE];
    #pragma unroll
    for (int n = 0; n < D_STATE; ++n) { st[n] = 0.f; P[n] = 1.f; }

    const int l0 = chunk * CLEN;
    for (int l = l0; l < l0 + CLEN; ++l) {
        int t = l * N_BATCH + b;
        float delta = softplusf(dtm[(size_t)t * D_INNER + d] + dtb);
        float u     = u_act[(size_t)t * D_INNER + d];
        float du    = delta * u;
        const float* bv = x_dbl + (size_t)t * XPROJ_N + DT_RANK;   // B[t][0..15]
        #pragma unroll
        for (int n = 0; n < D_STATE; ++n) {
            float dA = __expf(delta * Arow[n]);
            st[n] = st[n] * dA + du * bv[n];
            P[n] *= dA;
        }
    }
    size_t o = ((size_t)((b * NCHUNK + chunk) * D_INNER + d)) * D_STATE;
    #pragma unroll
    for (int n = 0; n < D_STATE; ++n) { stEnd[o + n] = st[n]; Ptot[o + n] = P[n]; }
}

// ---------------------------------------------------------------------------
// Scan pass 2: sequential combine of NCHUNK summaries per (b,d).
// stIn(chunk) = global state entering chunk; s' = stEnd + s * Ptot.
// ---------------------------------------------------------------------------
__global__ __launch_bounds__(256)
void scan_pass2(const float* __restrict__ stEnd, const float* __restrict__ Ptot,
                float* __restrict__ stIn)
{
    int idx = blockIdx.x * 256 + threadIdx.x;      // B * D_INNER = 4096
    int d = idx & (D_INNER - 1);
    int b = idx >> 11;
    float s[D_STATE];
    #pragma unroll
    for (int n = 0; n < D_STATE; ++n) s[n] = 0.f;
    for (int c = 0; c < NCHUNK; ++c) {
        size_t o = ((size_t)((b * NCHUNK + c) * D_INNER + d)) * D_STATE;
        #pragma unroll
        for (int n = 0; n < D_STATE; ++n) {
            stIn[o + n] = s[n];
            s[n] = stEnd[o + n] + s[n] * Ptot[o + n];
        }
    }
}

// ---------------------------------------------------------------------------
// Scan pass 3: rescan each chunk from its corrected initial state; emit
// y = C.h + D*u, gated by silu(z). Writes IN PLACE over u_act (each element
// read-then-written by the same thread; no other consumer remains).
// ---------------------------------------------------------------------------
__global__ __launch_bounds__(256)
void scan_pass3(const float* __restrict__ dtm, float* __restrict__ u_act,
                const float* __restrict__ x_dbl, const float* __restrict__ xz,
                const float* __restrict__ dt_bias, const float* __restrict__ A_log,
                const float* __restrict__ Dp, const float* __restrict__ stIn)
{
    int idx   = blockIdx.x * 256 + threadIdx.x;
    int d     = idx & (D_INNER - 1);
    int chunk = (idx >> 11) & (NCHUNK - 1);
    int b     = idx >> 15;

    float Arow[D_STATE];
    #pragma unroll
    for (int n = 0; n < D_STATE; ++n) Arow[n] = -__expf(A_log[d * D_STATE + n]);
    const float dtb = dt_bias[d];
    const float Dd  = Dp[d];

    float st[D_STATE];
    {
        size_t o = ((size_t)((b * NCHUNK + chunk) * D_INNER + d)) * D_STATE;
        #pragma unroll
        for (int n = 0; n < D_STATE; ++n) st[n] = stIn[o + n];
    }

    const int l0 = chunk * CLEN;
    for (int l = l0; l < l0 + CLEN; ++l) {
        int t = l * N_BATCH + b;
        float delta = softplusf(dtm[(size_t)t * D_INNER + d] + dtb);
        float u     = u_act[(size_t)t * D_INNER + d];
        float du    = delta * u;
        const float* bv = x_dbl + (size_t)t * XPROJ_N + DT_RANK;           // B
        const float* cv = bv + D_STATE;                                    // C
        float y = 0.f;
        #pragma unroll
        for (int n = 0; n < D_STATE; ++n) {
            float dA = __expf(delta * Arow[n]);
            st[n] = st[n] * dA + du * bv[n];
            y += st[n] * cv[n];
        }
        y += Dd * u;
        float z = xz[(size_t)t * (2 * D_INNER) + D_INNER + d];
        u_act[(size_t)t * D_INNER + d] = y * (z / (1.f + __expf(-z)));     // y * silu(z)
    }
}

// ---------------------------------------------------------------------------
extern "C" void kernel_launch(void* const* d_in, const int* in_sizes, int n_in,
                              void* d_out, int out_size, void* d_ws, size_t ws_size,
                              hipStream_t stream)
{
    (void)in_sizes; (void)n_in; (void)out_size; (void)ws_size;

    const float* hidden  = (const float*)d_in[0];  // [L,B,D_MODEL]
    const float* W_in    = (const float*)d_in[1];  // [2*D_INNER, D_MODEL]
    const float* conv_w  = (const float*)d_in[2];  // [D_INNER, 4]
    const float* conv_b  = (const float*)d_in[3];  // [D_INNER]
    const float* W_x     = (const float*)d_in[4];  // [96, D_INNER]
    const float* W_dt    = (const float*)d_in[5];  // [D_INNER, DT_RANK]
    const float* dt_bias = (const float*)d_in[6];  // [D_INNER]
    const float* A_log   = (const float*)d_in[7];  // [D_INNER, D_STATE]
    const float* Dp      = (const float*)d_in[8];  // [D_INNER]
    const float* W_out   = (const float*)d_in[9];  // [D_MODEL, D_INNER]
    float* out = (float*)d_out;                    // [L,B,D_MODEL] (rows t=l*B+b)

    // Workspace layout (floats)
    float* ws    = (float*)d_ws;
    float* xz    = ws;                                   // NTOK*4096 = 16,777,216
    float* x_act = xz    + (size_t)NTOK * 2 * D_INNER;   // NTOK*2048 =  8,388,608
    float* x_dbl = x_act + (size_t)NTOK * D_INNER;       // NTOK*96   =    393,216
    float* dtm   = x_dbl + (size_t)NTOK * XPROJ_N;       // NTOK*2048 =  8,388,608
    float* stEnd = dtm   + (size_t)NTOK * D_INNER;       // 2*16*2048*16 = 1,048,576
    float* Ptot  = stEnd + (size_t)N_BATCH * NCHUNK * D_INNER * D_STATE;
    float* stIn  = Ptot  + (size_t)N_BATCH * NCHUNK * D_INNER * D_STATE;

    dim3 blk(256);

    // 1) in_proj: xz[NTOK,4096] = hidden[NTOK,1024] @ W_in^T
    wmma_gemm_nt<<<dim3(2 * D_INNER / GBN, NTOK / GBM), blk, 0, stream>>>(
        hidden, W_in, xz, NTOK, 2 * D_INNER, D_MODEL, D_MODEL, D_MODEL, 2 * D_INNER);

    // 2) depthwise causal conv + SiLU -> x_act[NTOK, D_INNER]
    conv_silu_kernel<<<(NTOK * D_INNER) / 256, blk, 0, stream>>>(xz, conv_w, conv_b, x_act);

    // 3) x_proj: x_dbl[NTOK,96] = x_act @ W_x^T   (N tail guarded)
    wmma_gemm_nt<<<dim3((XPROJ_N + GBN - 1) / GBN, NTOK / GBM), blk, 0, stream>>>(
        x_act, W_x, x_dbl, NTOK, XPROJ_N, D_INNER, D_INNER, D_INNER, XPROJ_N);

    // 4) dt proj: dtm[NTOK, D_INNER] = dt_lr (x_dbl[:, :64]) @ W_dt^T
    wmma_gemm_nt<<<dim3(D_INNER / GBN, NTOK / GBM), blk, 0, stream>>>(
        x_dbl, W_dt, dtm, NTOK, D_INNER, DT_RANK, XPROJ_N, DT_RANK, D_INNER);

    // 5-7) chunked selective scan + D skip + silu(z) gate (in place into x_act)
    scan_pass1<<<(N_BATCH * D_INNER * NCHUNK) / 256, blk, 0, stream>>>(
        dtm, x_act, x_dbl, dt_bias, A_log, stEnd, Ptot);
    scan_pass2<<<(N_BATCH * D_INNER) / 256, blk, 0, stream>>>(stEnd, Ptot, stIn);
    scan_pass3<<<(N_BATCH * D_INNER * NCHUNK) / 256, blk, 0, stream>>>(
        dtm, x_act, x_dbl, xz, dt_bias, A_log, Dp, stIn);

    // 8) out_proj: out[NTOK, D_MODEL] = y_gated @ W_out^T
    wmma_gemm_nt<<<dim3(D_MODEL / GBN, NTOK / GBM), blk, 0, stream>>>(
        x_act, W_out, out, NTOK, D_MODEL, D_INNER, D_INNER, D_INNER, D_MODEL);
}